// VariableSelectionNetwork_206158430403
// MI455X (gfx1250) — compile-verified
//
#include <hip/hip_runtime.h>
#include <hip/hip_bf16.h>
#include <math.h>

// ---------------------------------------------------------------------------
// Fully-fused Variable Selection Network for gfx1250 (MI455X).
// B=32 T=512 F=16 D=64 H=128; 1024 blocks x 256 threads (8 wave32 / block).
// All intermediates for a 16-row slab stay in LDS (~173 KB of the 320 KB/WGP).
// GEMMs use V_WMMA_F32_16X16X4_F32 (f32 in/out -> bit-faithful precision;
// the kernel is HBM-bound at ~140MB / 23.3TB/s, so fp32 matrix rate is free).
// ---------------------------------------------------------------------------

typedef float v2f __attribute__((ext_vector_type(2)));
typedef float v8f __attribute__((ext_vector_type(8)));

#define ROWS_TOTAL 16384
#define F_ 16
#define D_ 64
#define H_ 128
#define FH_ 2048

// LDS strides (floats). All ≡ 4 (mod 16) so the 16-lane column gathers used to
// build WMMA A-fragments hit 16 distinct banks, and the two half-waves
// (reading k+0/1 vs k+2/3) land on disjoint bank groups.
#define ST_T 2052   // transformed [16][2048]
#define ST_X 68     // x tile      [16][64]
#define ST_P 132    // work tiles  [16][128]
#define ST_S 20     // small tiles [16][16]

#define SMEM_FLOATS (16*ST_T + 16*ST_X + 4*16*ST_P + 2*16*ST_S + 16*16)

__device__ __forceinline__ float rowsum16(float v) {
  v += __shfl_xor(v, 1, 16);
  v += __shfl_xor(v, 2, 16);
  v += __shfl_xor(v, 4, 16);
  v += __shfl_xor(v, 8, 16);
  return v;
}
__device__ __forceinline__ float rowmax16(float v) {
  v = fmaxf(v, __shfl_xor(v, 1, 16));
  v = fmaxf(v, __shfl_xor(v, 2, 16));
  v = fmaxf(v, __shfl_xor(v, 4, 16));
  v = fmaxf(v, __shfl_xor(v, 8, 16));
  return v;
}

// One 16x16 f32 output tile: A (16xK) from LDS, B (KxN row-major, ld=ldw)
// streamed from global (L2-resident weights), K-loop of wmma_f32_16x16x4.
// A layout (ISA 7.12.2): lanes 0-15 hold M=0..15; v0/v1 = K=k0,k0+1 for the
// low half-wave and K=k0+2,k0+3 for the high half-wave. B mirrors with N.
template <int K>
__device__ __forceinline__ v8f wmma_gemm(const float* __restrict__ As, int lda,
                                         const float* __restrict__ W, int ldw,
                                         int n0, int lane16, int hi) {
  v8f c = {0.f, 0.f, 0.f, 0.f, 0.f, 0.f, 0.f, 0.f};
  const float* arow = As + lane16 * lda + hi * 2;
  const float* wcol = W + (size_t)(hi * 2) * ldw + n0 + lane16;
#pragma unroll 1
  for (int kb = 0; kb < K; kb += 32) {
    if (kb + 32 < K) {  // pull next B chunk toward the WGP ahead of use
      __builtin_prefetch(wcol + (size_t)(kb + 32) * ldw, 0, 1);
      __builtin_prefetch(wcol + (size_t)(kb + 48) * ldw, 0, 1);
    }
#pragma unroll
    for (int k0 = kb; k0 < kb + 32; k0 += 4) {
      v2f a, b;
      a.x = arow[k0];
      a.y = arow[k0 + 1];
      b.x = wcol[(size_t)k0 * ldw];
      b.y = wcol[(size_t)(k0 + 1) * ldw];
      c = __builtin_amdgcn_wmma_f32_16x16x4_f32(false, a, false, b,
                                                (short)0, c, false, false);
    }
  }
  return c;
}

// C/D layout (ISA 7.12.2): VGPR v -> M = v (lanes 0-15) / v+8 (lanes 16-31),
// N = n0 + (lane & 15).
__device__ __forceinline__ void store_tile(float* dst, int ldd, int col0,
                                           v8f c, int lane16, int hi) {
#pragma unroll
  for (int v = 0; v < 8; ++v)
    dst[(v + hi * 8) * ldd + col0 + lane16] = c[v];
}

__global__ __launch_bounds__(256, 1)
void vsn_fused_206158430403(
    const float* __restrict__ x,
    const float* __restrict__ embW, const float* __restrict__ embB,
    const float* __restrict__ fc1W, const float* __restrict__ fc1B,
    const float* __restrict__ fc2W, const float* __restrict__ fc2B,
    const float* __restrict__ gW,   const float* __restrict__ gB,
    const float* __restrict__ lnG,  const float* __restrict__ lnB,
    const float* __restrict__ sf1W, const float* __restrict__ sf1B,
    const float* __restrict__ sf2W, const float* __restrict__ sf2B,
    const float* __restrict__ sgW,  const float* __restrict__ sgB,
    const float* __restrict__ sskW, const float* __restrict__ sskB,
    const float* __restrict__ slnG, const float* __restrict__ slnB,
    float* __restrict__ outSel, float* __restrict__ outW) {
  extern __shared__ float smem[];
  float* TRANS = smem;                    // [16][ST_T] transformed, f32
  float* XT = TRANS + 16 * ST_T;          // [16][ST_X] x tile
  float* P0 = XT + 16 * ST_X;             // emb tile
  float* P1 = P0 + 16 * ST_P;             // h1 tile
  float* P2 = P1 + 16 * ST_P;             // h2 tile
  float* P3 = P2 + 16 * ST_P;             // y tile (pre-LN)
  float* GP = P3 + 16 * ST_P;             // gate_pre 16x16
  float* SKT = GP + 16 * ST_S;            // skip 16x16
  float* WB = SKT + 16 * ST_S;            // softmax weights 16x16

  const int tid = threadIdx.x;
  const int wid = tid >> 5;
  const int lane = tid & 31;
  const int lane16 = lane & 15;
  const int hi = lane >> 4;
  const int n0 = wid * 16;                // this wave's N-tile
  const int lr = tid >> 4;                // pointwise row 0..15
  const int lc = tid & 15;                // pointwise col 0..15
  const int row0 = blockIdx.x * 16;

  // ---------------- Stage A: per-feature embed + GRN + LayerNorm ----------
  for (int f = 0; f < F_; ++f) {
    {  // cooperative x tile load: 16 rows x 64 floats (float4 per thread)
      const float* xp = x + (size_t)(row0 + lr) * (F_ * D_) + f * D_ + lc * 4;
      float4 v = *(const float4*)xp;
      float* dp = XT + lr * ST_X + lc * 4;
      dp[0] = v.x; dp[1] = v.y; dp[2] = v.z; dp[3] = v.w;
      if (f + 1 < F_) __builtin_prefetch(xp + D_, 0, 1);  // next feature
    }
    __syncthreads();

    // emb = x @ embW[f] + embB[f]
    v8f ce = wmma_gemm<D_>(XT, ST_X, embW + (size_t)f * D_ * H_, H_, n0, lane16, hi);
    {
      float b = embB[f * H_ + n0 + lane16];
#pragma unroll
      for (int v = 0; v < 8; ++v) ce[v] += b;
    }
    store_tile(P0, ST_P, n0, ce, lane16, hi);
    __syncthreads();

    // h1 = elu(emb @ fc1W[f] + b)
    v8f c1 = wmma_gemm<H_>(P0, ST_P, fc1W + (size_t)f * H_ * H_, H_, n0, lane16, hi);
    {
      float b = fc1B[f * H_ + n0 + lane16];
#pragma unroll
      for (int v = 0; v < 8; ++v) {
        float t = c1[v] + b;
        c1[v] = t > 0.f ? t : (expf(t) - 1.f);
      }
    }
    store_tile(P1, ST_P, n0, c1, lane16, hi);
    __syncthreads();

    // h2 = h1 @ fc2W[f] + b   (kept in regs for y, and in LDS as gate's A)
    v8f c2 = wmma_gemm<H_>(P1, ST_P, fc2W + (size_t)f * H_ * H_, H_, n0, lane16, hi);
    {
      float b = fc2B[f * H_ + n0 + lane16];
#pragma unroll
      for (int v = 0; v < 8; ++v) c2[v] += b;
    }
    store_tile(P2, ST_P, n0, c2, lane16, hi);
    __syncthreads();

    // gate = sigmoid(h2 @ gW[f] + b);  y = emb + gate * h2  (same tile coords)
    v8f cg = wmma_gemm<H_>(P2, ST_P, gW + (size_t)f * H_ * H_, H_, n0, lane16, hi);
    {
      float b = gB[f * H_ + n0 + lane16];
#pragma unroll
      for (int v = 0; v < 8; ++v) {
        float t = cg[v] + b;
        float s = 1.f / (1.f + expf(-t));
        cg[v] = ce[v] + s * c2[v];
      }
    }
    store_tile(P3, ST_P, n0, cg, lane16, hi);
    __syncthreads();

    // LayerNorm over H=128 per row -> TRANS[row][f*128 + h]
    {
      float vals[8];
      float s = 0.f, ss = 0.f;
#pragma unroll
      for (int j = 0; j < 8; ++j) {
        float v = P3[lr * ST_P + lc + 16 * j];
        vals[j] = v; s += v; ss += v * v;
      }
      s = rowsum16(s);
      ss = rowsum16(ss);
      float mu = s * (1.f / 128.f);
      float var = ss * (1.f / 128.f) - mu * mu;
      float rs = rsqrtf(var + 1e-5f);
#pragma unroll
      for (int j = 0; j < 8; ++j) {
        int h = lc + 16 * j;
        float nv = (vals[j] - mu) * rs * lnG[f * H_ + h] + lnB[f * H_ + h];
        TRANS[lr * ST_T + f * H_ + h] = nv;
      }
    }
    __syncthreads();
  }

  // ---------------- Stage B: softmax GRN over flattened [16][2048] --------
  // h1 = elu(flat @ s_fc1_W + b)   K = 2048
  v8f b1 = wmma_gemm<FH_>(TRANS, ST_T, sf1W, H_, n0, lane16, hi);
  {
    float b = sf1B[n0 + lane16];
#pragma unroll
    for (int v = 0; v < 8; ++v) {
      float t = b1[v] + b;
      b1[v] = t > 0.f ? t : (expf(t) - 1.f);
    }
  }
  store_tile(P1, ST_P, n0, b1, lane16, hi);
  __syncthreads();

  // h2 = h1 @ s_fc2_W + b
  v8f b2 = wmma_gemm<H_>(P1, ST_P, sf2W, H_, n0, lane16, hi);
  {
    float b = sf2B[n0 + lane16];
#pragma unroll
    for (int v = 0; v < 8; ++v) b2[v] += b;
  }
  store_tile(P2, ST_P, n0, b2, lane16, hi);
  __syncthreads();

  // Wave 0: gate_pre = h2 @ s_gate_W + b  (16x16, K=128)
  // Wave 1: skip     = flat @ s_skip_W + b (16x16, K=2048) -- concurrent
  if (wid == 0) {
    v8f g = wmma_gemm<H_>(P2, ST_P, sgW, F_, 0, lane16, hi);
    float b = sgB[lane16];
#pragma unroll
    for (int v = 0; v < 8; ++v) g[v] += b;
    store_tile(GP, ST_S, 0, g, lane16, hi);
  } else if (wid == 1) {
    v8f s = wmma_gemm<FH_>(TRANS, ST_T, sskW, F_, 0, lane16, hi);
    float b = sskB[lane16];
#pragma unroll
    for (int v = 0; v < 8; ++v) s[v] += b;
    store_tile(SKT, ST_S, 0, s, lane16, hi);
  }
  __syncthreads();

  // weights = softmax(LN(skip + sigmoid(gate_pre)*gate_pre))  (per row, F=16)
  {
    float gp = GP[lr * ST_S + lc];
    float sg = 1.f / (1.f + expf(-gp));
    float v = SKT[lr * ST_S + lc] + sg * gp;
    float s = rowsum16(v);
    float ss = rowsum16(v * v);
    float mu = s * (1.f / 16.f);
    float var = ss * (1.f / 16.f) - mu * mu;
    float nv = (v - mu) * rsqrtf(var + 1e-5f) * slnG[lc] + slnB[lc];
    float mx = rowmax16(nv);
    float e = expf(nv - mx);
    float se = rowsum16(e);
    float w = e / se;
    WB[lr * 16 + lc] = w;
    outW[(size_t)(row0 + lr) * F_ + lc] = w;
  }
  __syncthreads();

  // selected[row][h] = sum_f transformed[row][f*128+h] * weights[row][f]
  {
#pragma unroll
    for (int j = 0; j < 8; ++j) {
      int h = lc + 16 * j;
      float acc = 0.f;
#pragma unroll
      for (int f = 0; f < F_; ++f)
        acc += TRANS[lr * ST_T + f * H_ + h] * WB[lr * 16 + f];
      outSel[(size_t)(row0 + lr) * H_ + h] = acc;
    }
  }
}

extern "C" void kernel_launch(void* const* d_in, const int* in_sizes, int n_in,
                              void* d_out, int out_size, void* d_ws, size_t ws_size,
                              hipStream_t stream) {
  (void)in_sizes; (void)n_in; (void)out_size; (void)d_ws; (void)ws_size;
  const float* x    = (const float*)d_in[0];
  const float* embW = (const float*)d_in[1];
  const float* embB = (const float*)d_in[2];
  const float* fc1W = (const float*)d_in[3];
  const float* fc1B = (const float*)d_in[4];
  const float* fc2W = (const float*)d_in[5];
  const float* fc2B = (const float*)d_in[6];
  const float* gW   = (const float*)d_in[7];
  const float* gB   = (const float*)d_in[8];
  const float* lnG  = (const float*)d_in[9];
  const float* lnB  = (const float*)d_in[10];
  const float* sf1W = (const float*)d_in[11];
  const float* sf1B = (const float*)d_in[12];
  const float* sf2W = (const float*)d_in[13];
  const float* sf2B = (const float*)d_in[14];
  const float* sgW  = (const float*)d_in[15];
  const float* sgB  = (const float*)d_in[16];
  const float* sskW = (const float*)d_in[17];
  const float* sskB = (const float*)d_in[18];
  const float* slnG = (const float*)d_in[19];
  const float* slnB = (const float*)d_in[20];

  float* outSel = (float*)d_out;                       // [16384,128]
  float* outW   = outSel + (size_t)ROWS_TOTAL * H_;    // [16384,16]

  constexpr size_t shmem = (size_t)SMEM_FLOATS * sizeof(float);  // ~173 KB
  (void)hipFuncSetAttribute((const void*)vsn_fused_206158430403,
                            hipFuncAttributeMaxDynamicSharedMemorySize,
                            (int)shmem);

  vsn_fused_206158430403<<<ROWS_TOTAL / 16, 256, shmem, stream>>>(
      x, embW, embB, fc1W, fc1B, fc2W, fc2B, gW, gB, lnG, lnB,
      sf1W, sf1B, sf2W, sf2B, sgW, sgB, sskW, sskB, slnG, slnB,
      outSel, outW);
}